// CandidateSelector_12902081757597
// MI455X (gfx1250) — compile-verified
//
#include <hip/hip_runtime.h>
#include <hip/hip_bf16.h>

typedef __attribute__((ext_vector_type(2))) float v2f;
typedef __attribute__((ext_vector_type(4))) float v4f;
typedef __attribute__((ext_vector_type(8))) float v8f;

#define FEAT 256
#define EMB 64
#define TOPK 128
#define XT_STRIDE 264   // padded LDS row stride for 16x256 tile (floats)
#define W_STRIDE 66     // padded LDS row stride for 256x64 weights (floats)
#define WAVES 4
#define TPB (WAVES * 32)
#define CAP 1024

static __device__ __forceinline__ v8f wmma4(v2f a, v2f b, v8f c) {
  // V_WMMA_F32_16X16X4_F32: D = A(16x4 f32) * B(4x16 f32) + C(16x16 f32)
  return __builtin_amdgcn_wmma_f32_16x16x4_f32(false, a, false, b, (short)0, c,
                                               false, false);
}

// ---------------------------------------------------------------------------
// Kernel 1: h_T = mean(h[idx_targets], axis=0)   (64 floats)
// ---------------------------------------------------------------------------
__global__ void ht_mean_kernel(const float* __restrict__ h,
                               const int* __restrict__ idx,
                               float* __restrict__ hT, int nT) {
  const int d = threadIdx.x;  // 64 threads, one per embedding dim
  float s = 0.f;
  for (int t = 0; t < nT; ++t) s += h[(size_t)idx[t] * EMB + d];
  hT[d] = s / (float)nT;
}

// ---------------------------------------------------------------------------
// Kernel 2: fused gather + GEMM(256->64) + concat/relu + GEMM(256->64) + dot
// One wave handles one 16-row tile; weights cached in LDS per block.
// ---------------------------------------------------------------------------
__global__ __launch_bounds__(TPB) void fused_mlp_kernel(
    const float* __restrict__ x, const float* __restrict__ h,
    const float* __restrict__ degree, const float* __restrict__ beta,
    const int* __restrict__ exp_nodes,
    const float* __restrict__ W_raw, const float* __restrict__ b_raw,
    const float* __restrict__ W_num, const float* __restrict__ b_num,
    const float* __restrict__ W1, const float* __restrict__ b1,
    const float* __restrict__ W2, const float* __restrict__ b2,
    const float* __restrict__ hT, float* __restrict__ scores, int E) {
  extern __shared__ float lds[];
  float* sW0 = lds;                          // W_raw, [256][W_STRIDE]
  float* sW1 = lds + FEAT * W_STRIDE;        // W1,    [256][W_STRIDE]
  const int tid = threadIdx.x;
  const int wave = tid >> 5;
  const int lane = tid & 31;
  const int col = lane & 15;   // N / M index within 16
  const int hi = lane >> 4;    // half-wave select (K pair / M+8)
  float* sTile = lds + 2 * FEAT * W_STRIDE + wave * 16 * XT_STRIDE;

  // Stage both weight matrices into LDS once per block (coalesced).
  for (int i = tid; i < FEAT * EMB; i += TPB) {
    const int k = i >> 6, c = i & 63;
    sW0[k * W_STRIDE + c] = W_raw[i];
    sW1[k * W_STRIDE + c] = W1[i];
  }
  __syncthreads();

  // Per-lane constants (hoisted out of the tile loop).
  const float br0 = b_raw[col], br1 = b_raw[16 + col], br2 = b_raw[32 + col],
              br3 = b_raw[48 + col];
  const float b10 = b1[col], b11 = b1[16 + col], b12 = b1[32 + col],
              b13 = b1[48 + col];
  const float w20 = W2[col], w21 = W2[16 + col], w22 = W2[32 + col],
              w23 = W2[48 + col];
  const float b2s = b2[0];
  const float hta = fmaxf(hT[lane], 0.f), htb = fmaxf(hT[lane + 32], 0.f);
  const float wn0a = W_num[lane], wn0b = W_num[lane + 32];
  const float wn1a = W_num[64 + lane], wn1b = W_num[96 + lane];
  const float bna = b_num[lane], bnb = b_num[lane + 32];

  const int nTiles = (E + 15) >> 4;
  for (int t = blockIdx.x * WAVES + wave; t < nTiles; t += gridDim.x * WAVES) {
    const int base = t * 16;

    // ---- stage gathered X rows [16 x 256] into LDS (2x b128 per lane/row)
    for (int r = 0; r < 16; ++r) {
      int e = base + r;
      if (e >= E) e = E - 1;
      const float* src = x + (size_t)exp_nodes[e] * FEAT;
      float* dst = sTile + r * XT_STRIDE;
      *(v4f*)(dst + lane * 4) = *(const v4f*)(src + lane * 4);
      *(v4f*)(dst + 128 + lane * 4) = *(const v4f*)(src + 128 + lane * 4);
    }
    __asm__ volatile("s_wait_dscnt 0x0" ::: "memory");

    // ---- GEMM1: x_v = Xg @ W_raw  (K=256, 4 N-tiles of 16)
    v8f a0 = {0, 0, 0, 0, 0, 0, 0, 0}, a1 = a0, a2 = a0, a3 = a0;
    {
      const float* Ar = sTile + col * XT_STRIDE + 2 * hi;
#pragma unroll 4
      for (int k = 0; k < FEAT; k += 4) {
        v2f av;
        av.x = Ar[k];
        av.y = Ar[k + 1];
        const float* B0 = sW0 + (k + 2 * hi) * W_STRIDE;
        v2f bv;
        bv.x = B0[col];      bv.y = B0[W_STRIDE + col];      a0 = wmma4(av, bv, a0);
        bv.x = B0[16 + col]; bv.y = B0[W_STRIDE + 16 + col]; a1 = wmma4(av, bv, a1);
        bv.x = B0[32 + col]; bv.y = B0[W_STRIDE + 32 + col]; a2 = wmma4(av, bv, a2);
        bv.x = B0[48 + col]; bv.y = B0[W_STRIDE + 48 + col]; a3 = wmma4(av, bv, a3);
      }
    }

    // ---- assemble emb tile in-place (reuse X tile)
    // cols 0..63: relu(x_v + b_raw)  (C/D layout: vgpr j -> row j+8*hi, col)
#pragma unroll
    for (int j = 0; j < 8; ++j) {
      float* row = sTile + (j + 8 * hi) * XT_STRIDE;
      row[col] = fmaxf(a0[j] + br0, 0.f);
      row[16 + col] = fmaxf(a1[j] + br1, 0.f);
      row[32 + col] = fmaxf(a2[j] + br2, 0.f);
      row[48 + col] = fmaxf(a3[j] + br3, 0.f);
    }
    // cols 64..127: relu(h_v); 128..191: relu(h_T); 192..255: relu(emb_num)
    for (int r = 0; r < 16; ++r) {
      int e = base + r;
      if (e >= E) e = E - 1;
      const int node = exp_nodes[e];
      const float* hv = h + (size_t)node * EMB;
      float* row = sTile + r * XT_STRIDE;
      row[64 + lane] = fmaxf(hv[lane], 0.f);
      row[96 + lane] = fmaxf(hv[32 + lane], 0.f);
      row[128 + lane] = hta;
      row[160 + lane] = htb;
      const float dg = degree[node], bt = beta[node];
      row[192 + lane] = fmaxf(dg * wn0a + bt * wn1a + bna, 0.f);
      row[224 + lane] = fmaxf(dg * wn0b + bt * wn1b + bnb, 0.f);
    }
    __asm__ volatile("s_wait_dscnt 0x0" ::: "memory");

    // ---- GEMM2: hidden = emb @ W1
    v8f c0 = {0, 0, 0, 0, 0, 0, 0, 0}, c1 = c0, c2 = c0, c3 = c0;
    {
      const float* Ar = sTile + col * XT_STRIDE + 2 * hi;
#pragma unroll 4
      for (int k = 0; k < FEAT; k += 4) {
        v2f av;
        av.x = Ar[k];
        av.y = Ar[k + 1];
        const float* B0 = sW1 + (k + 2 * hi) * W_STRIDE;
        v2f bv;
        bv.x = B0[col];      bv.y = B0[W_STRIDE + col];      c0 = wmma4(av, bv, c0);
        bv.x = B0[16 + col]; bv.y = B0[W_STRIDE + 16 + col]; c1 = wmma4(av, bv, c1);
        bv.x = B0[32 + col]; bv.y = B0[W_STRIDE + 32 + col]; c2 = wmma4(av, bv, c2);
        bv.x = B0[48 + col]; bv.y = B0[W_STRIDE + 48 + col]; c3 = wmma4(av, bv, c3);
      }
    }

    // ---- scores = relu(hidden + b1) @ W2 + b2 (16-lane shuffle reduction)
#pragma unroll
    for (int j = 0; j < 8; ++j) {
      float s = fmaxf(c0[j] + b10, 0.f) * w20 +
                fmaxf(c1[j] + b11, 0.f) * w21 +
                fmaxf(c2[j] + b12, 0.f) * w22 +
                fmaxf(c3[j] + b13, 0.f) * w23;
      s += __shfl_xor(s, 1, 32);
      s += __shfl_xor(s, 2, 32);
      s += __shfl_xor(s, 4, 32);
      s += __shfl_xor(s, 8, 32);
      if (col == 0) {
        const int e = base + j + 8 * hi;
        if (e < E) scores[e] = s + b2s;
      }
    }
  }
}

// ---------------------------------------------------------------------------
// Kernel 3: exact top-128 (softmax is monotonic -> top-k on raw scores).
// Radix threshold search on order-preserving keys, then rank placement with
// JAX tie rule (value desc, index asc). out[0..127]=1.0, out[128..255]=idx.
// ---------------------------------------------------------------------------
__global__ __launch_bounds__(1024) void topk_kernel(
    const float* __restrict__ scores, const int* __restrict__ exp_nodes,
    float* __restrict__ out, int E, int K) {
  __shared__ unsigned s_cnt;
  __shared__ unsigned s_key[CAP];
  __shared__ int s_id[CAP];
  const int tid = threadIdx.x;
  const int nthr = blockDim.x;

  // 32-step binary search for the K-th largest key.
  unsigned T = 0u;
  for (int bit = 31; bit >= 0; --bit) {
    const unsigned candv = T | (1u << bit);
    if (tid == 0) s_cnt = 0;
    __syncthreads();
    unsigned local = 0;
    for (int e = tid; e < E; e += nthr) {
      const unsigned u = __float_as_uint(scores[e]);
      const unsigned k = (u & 0x80000000u) ? ~u : (u | 0x80000000u);
      if (k >= candv) ++local;
    }
    atomicAdd(&s_cnt, local);
    __syncthreads();
    if (s_cnt >= (unsigned)K) T = candv;
    __syncthreads();
  }

  // Collect all candidates with key >= threshold (>= K of them, capped).
  if (tid == 0) s_cnt = 0;
  __syncthreads();
  for (int e = tid; e < E; e += nthr) {
    const unsigned u = __float_as_uint(scores[e]);
    const unsigned k = (u & 0x80000000u) ? ~u : (u | 0x80000000u);
    if (k >= T) {
      const unsigned p = atomicAdd(&s_cnt, 1u);
      if (p < CAP) {
        s_key[p] = k;
        s_id[p] = e;
      }
    }
  }
  __syncthreads();
  const int n = (int)(s_cnt < CAP ? s_cnt : CAP);

  // Rank placement: strict total order -> distinct ranks 0..n-1.
  for (int i = tid; i < n; i += nthr) {
    const unsigned ki = s_key[i];
    const int ei = s_id[i];
    int r = 0;
    for (int j = 0; j < n; ++j) {
      const unsigned kj = s_key[j];
      const int ej = s_id[j];
      r += (kj > ki) || (kj == ki && ej < ei);
    }
    if (r < K) {
      out[r] = 1.0f;                       // straight-through forward value
      out[K + r] = (float)exp_nodes[ei];   // cand_indices
    }
  }
}

// ---------------------------------------------------------------------------
extern "C" void kernel_launch(void* const* d_in, const int* in_sizes, int n_in,
                              void* d_out, int out_size, void* d_ws,
                              size_t ws_size, hipStream_t stream) {
  const float* x = (const float*)d_in[0];
  const float* h = (const float*)d_in[1];
  const float* degree = (const float*)d_in[2];
  const float* beta = (const float*)d_in[3];
  const int* exp_nodes = (const int*)d_in[4];
  const int* idx_targets = (const int*)d_in[5];
  const float* W_raw = (const float*)d_in[6];
  const float* b_raw = (const float*)d_in[7];
  const float* W_num = (const float*)d_in[8];
  const float* b_num = (const float*)d_in[9];
  const float* W1 = (const float*)d_in[10];
  const float* b1 = (const float*)d_in[11];
  const float* W2 = (const float*)d_in[12];
  const float* b2 = (const float*)d_in[13];
  const int E = in_sizes[4];
  const int nT = in_sizes[5];

  float* hT = (float*)d_ws;       // [64]
  float* scores = hT + 64;        // [E]
  float* out = (float*)d_out;     // [256] = candidates(128) ++ indices(128)

  ht_mean_kernel<<<1, EMB, 0, stream>>>(h, idx_targets, hT, nT);

  const size_t shbytes =
      (size_t)(2 * FEAT * W_STRIDE + WAVES * 16 * XT_STRIDE) * sizeof(float);
  (void)hipFuncSetAttribute((const void*)fused_mlp_kernel,
                            hipFuncAttributeMaxDynamicSharedMemorySize,
                            (int)shbytes);
  fused_mlp_kernel<<<256, TPB, shbytes, stream>>>(
      x, h, degree, beta, exp_nodes, W_raw, b_raw, W_num, b_num, W1, b1, W2,
      b2, hT, scores, E);

  topk_kernel<<<1, 1024, 0, stream>>>(scores, exp_nodes, out, E, TOPK);
}